// diceLoss_41274635715148
// MI455X (gfx1250) — compile-verified
//
#include <hip/hip_runtime.h>

// ---------------------------------------------------------------------------
// Dice loss (2-class softmax) for logits[16,2,1024,1024] f32, labels[16,1,1024,1024] i32.
// Bandwidth-bound streaming reduction: ~201 MB @ 23.3 TB/s => ~8.6 us floor.
// Stage 1: per-block partial sums (NT b128 loads, exp2/rcp softmax, WMMA f32
//          16x16x4 ones-matrix reduction on the matrix pipe).
// Stage 2: fixed-order deterministic combine + final scalar.
// ---------------------------------------------------------------------------

typedef float v4f __attribute__((ext_vector_type(4)));
typedef int   v4i __attribute__((ext_vector_type(4)));
typedef float v2f __attribute__((ext_vector_type(2)));
typedef float v8f __attribute__((ext_vector_type(8)));

#define HW_PIX            (1024 * 1024)
#define NBATCH            16
#define BLOCKS_PER_BATCH  128
#define PIX_PER_BLOCK     (HW_PIX / BLOCKS_PER_BATCH)        // 8192
#define NTHREADS          256                                 // 8 wave32 waves
#define PIX_PER_ITER      (NTHREADS * 4)                      // 1024
#define NITERS            (PIX_PER_BLOCK / PIX_PER_ITER)      // 8
#define LOG2E             1.44269504088896340736f

// Stage 1: each block reduces an 8192-pixel slice of one batch plane into
// 5 partial sums: {s0, s1, q0, q1, cnt1} stored at ws[blockIdx.x*8 + j].
__global__ __launch_bounds__(NTHREADS) void dice_partial_kernel(
    const float* __restrict__ logits,   // [16][2][1024][1024]
    const int*   __restrict__ labels,   // [16][1][1024][1024]
    float*       __restrict__ ws)       // [2048][8]
{
  const int tid   = threadIdx.x;
  const int batch = blockIdx.x / BLOCKS_PER_BATCH;
  const int blk   = blockIdx.x % BLOCKS_PER_BATCH;

  const float* l0p  = logits + (size_t)batch * 2 * HW_PIX;
  const float* l1p  = l0p + HW_PIX;
  const int*   labp = labels + (size_t)batch * HW_PIX;
  const int pixBase = blk * PIX_PER_BLOCK + tid * 4;

  float s0 = 0.f, s1 = 0.f, q0 = 0.f, q1 = 0.f, c1 = 0.f;

#pragma unroll
  for (int i = 0; i < NITERS; ++i) {
    const int idx = pixBase + i * PIX_PER_ITER;
    // Single-use streaming data: non-temporal b128 loads (skip cache rinse).
    v4f L0 = __builtin_nontemporal_load((const v4f*)(l0p + idx));
    v4f L1 = __builtin_nontemporal_load((const v4f*)(l1p + idx));
    v4i LB = __builtin_nontemporal_load((const v4i*)(labp + idx));
    // Speculative prefetch ~8KB ahead (global_prefetch_b8).
    __builtin_prefetch(l0p + idx + 2 * PIX_PER_ITER, 0, 0);
    __builtin_prefetch(l1p + idx + 2 * PIX_PER_ITER, 0, 0);

#pragma unroll
    for (int e = 0; e < 4; ++e) {
      float d  = (L1[e] - L0[e]) * LOG2E;
      float ex = __builtin_amdgcn_exp2f(d);          // v_exp_f32
      float p0 = __builtin_amdgcn_rcpf(1.0f + ex);   // v_rcp_f32; rcp(inf)=0 -> safe
      float p1 = 1.0f - p0;
      float tf = (float)LB[e];                       // label in {0,1}; t^2 == t
      s0 += (1.0f - tf) * p0;
      s1 += tf * p1;
      q0 += p0 * p0;
      q1 += p1 * p1;
      c1 += tf;
    }
  }

  // Stage the 256x5 per-thread partials in LDS, accumulator-major.
  __shared__ float red[5 * NTHREADS];
  red[0 * NTHREADS + tid] = s0;
  red[1 * NTHREADS + tid] = s1;
  red[2 * NTHREADS + tid] = q0;
  red[3 * NTHREADS + tid] = q1;
  red[4 * NTHREADS + tid] = c1;
  __syncthreads();

  // Wave 0 (all 32 lanes active -> EXEC all-1s, as WMMA requires) reduces each
  // 256-value array with a chain of V_WMMA_F32_16X16X4_F32 against B = ones:
  // D[m][n] = sum_k A[m][k] + C[m][n]  => 64 values folded per WMMA.
  if (tid < 32) {
    const int m  = tid & 15;          // A-matrix row held by this lane
    const int kb = (tid >> 4) << 1;   // lanes 0-15: K={0,1}; lanes 16-31: K={2,3}
    const v2f bones = {1.0f, 1.0f};   // all-ones 4x16 B matrix

    for (int j = 0; j < 5; ++j) {
      v8f acc = {};
#pragma unroll
      for (int c = 0; c < 4; ++c) {
        const float* base = &red[j * NTHREADS + c * 64 + m * 4 + kb];
        v2f a;
        a.x = base[0];
        a.y = base[1];
        acc = __builtin_amdgcn_wmma_f32_16x16x4_f32(
            /*neg_a=*/false, a, /*neg_b=*/false, bones,
            /*c_mod=*/(short)0, acc, /*reuse_a=*/false, /*reuse_b=*/false);
      }
      // Each lane holds 8 row-sums (rows m..m+? per VGPR); all columns equal.
      float part = acc[0] + acc[1] + acc[2] + acc[3] +
                   acc[4] + acc[5] + acc[6] + acc[7];
      part += __shfl_xor(part, 16, 32);  // rows 0-7 half + rows 8-15 half
      if (tid == 0) ws[(size_t)blockIdx.x * 8 + j] = part;  // plain store: deterministic
    }
  }
}

// Stage 2: 32 lanes, lane = batch*2 + class. Fixed-order sum over the 128
// block-partials per batch, then dice + mean + write scalar.
__global__ __launch_bounds__(32) void dice_final_kernel(
    const float* __restrict__ ws, float* __restrict__ out)
{
  const int lane = threadIdx.x;       // 0..31
  const int b = lane >> 1;
  const int c = lane & 1;

  float s = 0.f, q = 0.f, cnt = 0.f;
  const float* base = ws + (size_t)b * BLOCKS_PER_BATCH * 8;
  for (int k = 0; k < BLOCKS_PER_BATCH; ++k) {
    const float* w = base + (size_t)k * 8;
    s   += w[c];
    q   += w[2 + c];
    cnt += w[4];
  }
  const float n    = c ? cnt : ((float)HW_PIX - cnt);  // sum(t_c^2)
  float dice = (2.0f * s) / (q + n + 1e-6f);

  // wave32 tree reduction of the 32 (b,c) dice terms
  for (int off = 16; off; off >>= 1) dice += __shfl_xor(dice, off, 32);
  if (lane == 0) out[0] = 1.0f - dice * (1.0f / 32.0f);
}

extern "C" void kernel_launch(void* const* d_in, const int* in_sizes, int n_in,
                              void* d_out, int out_size, void* d_ws, size_t ws_size,
                              hipStream_t stream) {
  const float* logits = (const float*)d_in[0];
  const int*   labels = (const int*)d_in[1];
  float*       ws     = (float*)d_ws;   // needs 2048*8*4 = 64 KB, fully rewritten each call
  float*       out    = (float*)d_out;

  dice_partial_kernel<<<dim3(NBATCH * BLOCKS_PER_BATCH), dim3(NTHREADS), 0, stream>>>(
      logits, labels, ws);
  dice_final_kernel<<<dim3(1), dim3(32), 0, stream>>>(ws, out);
}